// RNN_35081292874267
// MI455X (gfx1250) — compile-verified
//
#include <hip/hip_runtime.h>

// CDNA5 (gfx1250) RNN:  out[n,t,:] = tanh(x[n,t,:]@Wx + b + h_{t-1}@Wh)
//   Kernel 1: (N*T,D)x(D,H)+b -> d_out (doubles as xw buffer). Each wave owns a
//             64x64 macro-tile (4x4 register blocking, 16 f32 accumulators):
//             per K-step, 4 A-frags x 4 B-frags feed 16 WMMAs -> 4x reuse on
//             BOTH operands (~2 GB L2 traffic vs ~5 GB unblocked).
//   Kernel 2: per-timestep GEMM, slice t-1 read as h_prev, slice t updated in
//             place. K=1024 split over 4 waves (LDS combine) + dual accumulators
//             => 32-deep WMMA chains; serial phase is latency-bound and Wh /
//             h_prev are L2-resident (192 MB L2 vs 4.25 MB working set).

typedef __attribute__((ext_vector_type(2))) float v2f;
typedef __attribute__((ext_vector_type(8))) float v8f;

static constexpr int kN = 64, kT = 512, kD = 512, kH = 1024;

__device__ __forceinline__ v8f wmma_f32(v2f a, v2f b, v8f c) {
  // 8 args: (neg_a, A, neg_b, B, c_mod, C, reuse_a, reuse_b)
  return __builtin_amdgcn_wmma_f32_16x16x4_f32(false, a, false, b, (short)0, c,
                                               false, false);
}

// ---------------------------------------------------------------------------
// Kernel 1: XW = X @ Wx + b.   X:(N*T,D), Wx:(D,H), out:(N*T,H), row-major.
// Block = 128 threads = 4 waves. Wave w covers rows [row0,row0+64) x cols
// [blockIdx.y*256 + 64w, +64): 4 M-subtiles x 4 N-subtiles, 16 accumulators.
// Grid = (N*T/64, H/256) = (512, 4).
// ---------------------------------------------------------------------------
__global__ __launch_bounds__(128) void xw_gemm_kernel(
    const float* __restrict__ X, const float* __restrict__ Wx,
    const float* __restrict__ bias, float* __restrict__ out) {
  const int lane    = threadIdx.x & 31;
  const int wave    = threadIdx.x >> 5;
  const int m       = lane & 15;
  const int halfsel = lane >> 4;  // 0: lane carries K {0,1}; 1: K {2,3}
  const int row0 = blockIdx.x * 64;
  const int col0 = blockIdx.y * 256 + wave * 64;  // wave's 64-col strip

  const float* arow = X + (size_t)(row0 + m) * kD + 2 * halfsel;
  const float* bcol = Wx + (size_t)(2 * halfsel) * kH + col0 + m;

  v8f acc[4][4] = {};  // [M-subtile][N-subtile]
#pragma unroll 2
  for (int k = 0; k < kD; k += 4) {
    // WGP-scope prefetch of the strided B row stream a few K-groups ahead.
    __builtin_prefetch(bcol + (size_t)(k + 16) * kH, 0, 3);
    v2f a[4], b[4];
#pragma unroll
    for (int s = 0; s < 4; ++s)  // contiguous b64 loads
      a[s] = *(const v2f*)(arow + (size_t)s * 16 * kD + k);
#pragma unroll
    for (int c = 0; c < 4; ++c) {
      b[c].x = bcol[(size_t)k * kH + c * 16];        // Wx[k+2h][col0+16c+m]
      b[c].y = bcol[(size_t)(k + 1) * kH + c * 16];  // Wx[k+2h+1][...]
    }
#pragma unroll
    for (int s = 0; s < 4; ++s)
#pragma unroll
      for (int c = 0; c < 4; ++c) acc[s][c] = wmma_f32(a[s], b[c], acc[s][c]);
  }

#pragma unroll
  for (int c = 0; c < 4; ++c) {
    const int col = col0 + c * 16 + m;
    const float bb = bias[col];
#pragma unroll
    for (int s = 0; s < 4; ++s)
#pragma unroll
      for (int r = 0; r < 8; ++r) {
        const int row = row0 + s * 16 + r + 8 * halfsel;  // C: VGPR r -> M=r/r+8
        out[(size_t)row * kH + col] = acc[s][c][r] + bb;
      }
  }
}

// ---------------------------------------------------------------------------
// Kernel 2: one recurrence step.  yt (in: xw_t, out: h_t) is a strided (N,H)
// slice of d_out; hprev is slice t-1 (or h0).  Block = 128 threads = 4 waves;
// all 4 waves compute the SAME 16x16 tile over a quarter of K each, partials
// combined through LDS -> 64 WMMAs per wave, 32-deep dependency chains.
// Grid = (N/16, H/16) = (4, 64).
// ---------------------------------------------------------------------------
__global__ __launch_bounds__(128) void rnn_step_kernel(
    const float* __restrict__ hprev, long hstride,
    const float* __restrict__ Wh, float* __restrict__ yt, long ystride) {
  __shared__ float red[3][8][32];

  const int lane    = threadIdx.x & 31;
  const int wave    = threadIdx.x >> 5;  // 0..3: K-quarter
  const int m       = lane & 15;
  const int halfsel = lane >> 4;
  const int row0 = blockIdx.x * 16;
  const int col  = blockIdx.y * 16 + m;

  const int k0 = wave * (kH / 4);
  const float* arow = hprev + (size_t)(row0 + m) * hstride + k0 + 2 * halfsel;
  const float* bcol = Wh + (size_t)(k0 + 2 * halfsel) * kH + col;

  v8f acc0 = {};
  v8f acc1 = {};
#pragma unroll 4
  for (int k = 0; k < kH / 4; k += 8) {
    __builtin_prefetch(bcol + (size_t)(k + 16) * kH, 0, 3);
    v2f a0 = *(const v2f*)(arow + k);
    v2f a1 = *(const v2f*)(arow + k + 4);
    v2f b0, b1;
    b0.x = bcol[(size_t)k * kH];
    b0.y = bcol[(size_t)(k + 1) * kH];
    b1.x = bcol[(size_t)(k + 4) * kH];
    b1.y = bcol[(size_t)(k + 5) * kH];
    acc0 = wmma_f32(a0, b0, acc0);
    acc1 = wmma_f32(a1, b1, acc1);
  }
  v8f acc = acc0 + acc1;

  if (wave != 0) {
#pragma unroll
    for (int r = 0; r < 8; ++r) red[wave - 1][r][lane] = acc[r];
  }
  __syncthreads();
  if (wave == 0) {
#pragma unroll
    for (int r = 0; r < 8; ++r) {
      const int row = row0 + r + 8 * halfsel;
      float* p = yt + (size_t)row * ystride + col;
      const float v = *p + acc[r] + red[0][r][lane] + red[1][r][lane] +
                      red[2][r][lane];
      *p = tanhf(v);
    }
  }
}

// ---------------------------------------------------------------------------
extern "C" void kernel_launch(void* const* d_in, const int* in_sizes, int n_in,
                              void* d_out, int out_size, void* d_ws,
                              size_t ws_size, hipStream_t stream) {
  const float* x  = (const float*)d_in[0];  // (N,T,D)
  const float* h0 = (const float*)d_in[1];  // (N,H)
  const float* Wx = (const float*)d_in[2];  // (D,H)
  const float* Wh = (const float*)d_in[3];  // (H,H)
  const float* b  = (const float*)d_in[4];  // (H)
  float* out = (float*)d_out;               // (N,T,H), doubles as xw buffer

  dim3 g1((kN * kT) / 64, kH / 256);
  xw_gemm_kernel<<<g1, 128, 0, stream>>>(x, Wx, b, out);

  dim3 g2(kN / 16, kH / 16);
  const long ys = (long)kT * kH;  // row (n) stride of a time slice of out
  for (int t = 0; t < kT; ++t) {
    const float* hp = (t == 0) ? h0 : (out + (size_t)(t - 1) * kH);
    const long hs   = (t == 0) ? (long)kH : ys;
    rnn_step_kernel<<<g2, 128, 0, stream>>>(hp, hs, Wh, out + (size_t)t * kH,
                                            ys);
  }
}